// RcModel_11390253269451
// MI455X (gfx1250) — compile-verified
//
#include <hip/hip_runtime.h>
#include <hip/hip_bf16.h>
#include <math.h>

// ---------------------------------------------------------------------------
// Problem constants (from the reference)
// ---------------------------------------------------------------------------
#define V_  50000
#define D_  300
#define E_  50
#define DE_ 100
#define HW_ 256
#define HE_ 64
#define H_  512
#define B_  256
#define S_  512
#define CAT_ (3 * HW_ + HE_)   // 832
#define NEG_ (-1e30f)

typedef __attribute__((ext_vector_type(2))) float v2f;
typedef __attribute__((ext_vector_type(8))) float v8f;

// ---------------------------------------------------------------------------
// Kernel 1: embedding gather + 3 masked segment-maxes in a single pass.
// Block = one batch row. Each active thread owns a 16B-aligned float4 chunk of
// the 300-dim embedding (rows are 1200 B = 16B aligned), so gathers lower to
// global_load_b128 — 4x fewer VMEM ops than scalar. The mask value is uniform
// across lanes (depends only on s), so the branch is a cheap scalar branch and
// only one v_max_f32 per element executes.
// ---------------------------------------------------------------------------
__global__ void embed_segmax(const int* __restrict__ inputs,
                             const int* __restrict__ masks,
                             const float* __restrict__ Wword,
                             float* __restrict__ segA /* [3*B_, D_] */) {
    const int b = blockIdx.x;
    __shared__ int s_idx[S_];
    __shared__ int s_msk[S_];
    for (int t = threadIdx.x; t < S_; t += blockDim.x) {
        s_idx[t] = inputs[b * S_ + t];
        s_msk[t] = masks[b * S_ + t];
    }
    __syncthreads();
    const int c = threadIdx.x;          // float4 chunk id
    if (c < D_ / 4) {                   // 75 chunks
        const int d = c * 4;
        float4 m1 = make_float4(NEG_, NEG_, NEG_, NEG_);
        float4 m2 = m1, m3 = m1;
#pragma unroll 4
        for (int s = 0; s < S_; ++s) {
            const float4 v = *(const float4*)(Wword + (size_t)s_idx[s] * D_ + d);
            const int mk = s_msk[s];    // lane-uniform -> scalar branch
            if (mk == 1) {
                m1.x = fmaxf(m1.x, v.x); m1.y = fmaxf(m1.y, v.y);
                m1.z = fmaxf(m1.z, v.z); m1.w = fmaxf(m1.w, v.w);
            } else if (mk == 2) {
                m2.x = fmaxf(m2.x, v.x); m2.y = fmaxf(m2.y, v.y);
                m2.z = fmaxf(m2.z, v.z); m2.w = fmaxf(m2.w, v.w);
            } else if (mk == 3) {
                m3.x = fmaxf(m3.x, v.x); m3.y = fmaxf(m3.y, v.y);
                m3.z = fmaxf(m3.z, v.z); m3.w = fmaxf(m3.w, v.w);
            }
        }
        // segA rows are 1200 B (16B aligned), d is a multiple of 4 -> b128 stores
        *(float4*)(segA + (size_t)(0 * B_ + b) * D_ + d) = m1;
        *(float4*)(segA + (size_t)(1 * B_ + b) * D_ + d) = m2;
        *(float4*)(segA + (size_t)(2 * B_ + b) * D_ + d) = m3;
    }
}

// ---------------------------------------------------------------------------
// Kernel 2: entity embedding gather -> entA [B_, DE_]
// ---------------------------------------------------------------------------
__global__ void ent_gather(const int* __restrict__ entities,
                           const float* __restrict__ Went,
                           float* __restrict__ entA) {
    const int b = blockIdx.x;
    const int d = threadIdx.x;
    if (d < DE_) entA[b * DE_ + d] = Went[(size_t)entities[b] * DE_ + d];
}

// ---------------------------------------------------------------------------
// WMMA fp32 GEMM via V_WMMA_F32_16X16X4_F32, one 16x32 output tile per wave
// (two accumulators sharing one A fragment -> halves A traffic, doubles WMMA
// density per loop iteration).
// A layout (ISA 16x4 f32): lane<16 -> K={0,1}, lane>=16 -> K={2,3}; M = lane%16.
// B layout (row-across-lanes): N = lane%16, same K split across the half-wave.
// C/D: element i -> row = i + 8*(lane/16), col = lane%16.
// Epilogue: + bias, exact GELU (erf), mode-specific store:
//   MODE 0: stacked [3*B_,HW_] word path -> scatter into features[b, m*HW_ + n]
//   MODE 1: entity path [B_,HE_]        -> features[b, 3*HW_ + n]
//   MODE 2: combine path [B_,H_]        -> hidden[b, n]
// ---------------------------------------------------------------------------
template <int MODE>
__global__ void gemm_wmma_gelu(const float* __restrict__ A, int lda,
                               const float* __restrict__ B, int ldb,
                               const float* __restrict__ bias,
                               float* __restrict__ out, int K) {
    const int lane = threadIdx.x & 31;
    const int half = lane >> 4;
    const int lm   = lane & 15;
    const int tileN = blockIdx.x * 32;
    const int tileM = blockIdx.y * 16;

    v8f c0 = {}, c1 = {};
    const float* __restrict__ Arow  = A + (size_t)(tileM + lm) * lda;
    const float* __restrict__ Bcol0 = B + tileN + lm;
    const float* __restrict__ Bcol1 = Bcol0 + 16;

    for (int k = 0; k < K; k += 4) {
        const int kk = k + 2 * half;
        v2f a, b0, b1;
        a.x  = Arow[kk];
        a.y  = Arow[kk + 1];
        b0.x = Bcol0[(size_t)kk * ldb];
        b0.y = Bcol0[(size_t)(kk + 1) * ldb];
        b1.x = Bcol1[(size_t)kk * ldb];
        b1.y = Bcol1[(size_t)(kk + 1) * ldb];
        c0 = __builtin_amdgcn_wmma_f32_16x16x4_f32(false, a, false, b0,
                                                   (short)0, c0, false, false);
        c1 = __builtin_amdgcn_wmma_f32_16x16x4_f32(false, a, false, b1,
                                                   (short)0, c1, false, false);
    }

#pragma unroll
    for (int t = 0; t < 2; ++t) {
        const v8f& c = (t == 0) ? c0 : c1;
        const int n  = tileN + t * 16 + lm;
        const float bv = bias[n];
#pragma unroll
        for (int i = 0; i < 8; ++i) {
            const int r = tileM + i + 8 * half;
            const float x = c[i] + bv;
            const float g = 0.5f * x * (1.0f + erff(x * 0.70710678118654752f));
            if (MODE == 0) {
                const int m  = r >> 8;      // which segment (0..2)
                const int bb = r & (B_ - 1);
                out[(size_t)bb * CAT_ + m * HW_ + n] = g;
            } else if (MODE == 1) {
                out[(size_t)r * CAT_ + 3 * HW_ + n] = g;
            } else {
                out[(size_t)r * H_ + n] = g;
            }
        }
    }
}

// ---------------------------------------------------------------------------
// Kernel 4: classifier (512 -> 2), log-softmax, per-sample NLL.
// One 64-thread block per batch row.
// ---------------------------------------------------------------------------
__global__ void cls_loss(const float* __restrict__ hidden,
                         const float* __restrict__ clsW,
                         const float* __restrict__ clsb,
                         const int* __restrict__ labels,
                         float* __restrict__ out,
                         float* __restrict__ losses) {
    const int b = blockIdx.x;
    const int t = threadIdx.x;
    __shared__ float r0[64], r1[64];
    float a0 = 0.f, a1 = 0.f;
    for (int j = t; j < H_; j += 64) {
        const float h = hidden[(size_t)b * H_ + j];
        a0 += h * clsW[j * 2 + 0];
        a1 += h * clsW[j * 2 + 1];
    }
    r0[t] = a0; r1[t] = a1;
    __syncthreads();
    for (int s = 32; s > 0; s >>= 1) {
        if (t < s) { r0[t] += r0[t + s]; r1[t] += r1[t + s]; }
        __syncthreads();
    }
    if (t == 0) {
        const float l0 = r0[0] + clsb[0];
        const float l1 = r1[0] + clsb[1];
        out[b * 2 + 0] = l0;
        out[b * 2 + 1] = l1;
        const float mx  = fmaxf(l0, l1);
        const float lse = mx + logf(expf(l0 - mx) + expf(l1 - mx));
        losses[b] = lse - ((labels[b] == 0) ? l0 : l1);
    }
}

// ---------------------------------------------------------------------------
// Kernel 5: mean of per-sample losses -> d_out[512]
// ---------------------------------------------------------------------------
__global__ void loss_mean(const float* __restrict__ losses, float* __restrict__ out) {
    __shared__ float r[B_];
    const int t = threadIdx.x;
    r[t] = losses[t];
    __syncthreads();
    for (int s = B_ / 2; s > 0; s >>= 1) {
        if (t < s) r[t] += r[t + s];
        __syncthreads();
    }
    if (t == 0) out[B_ * 2] = r[0] * (1.0f / B_);
}

// ---------------------------------------------------------------------------
// Host launch
// ---------------------------------------------------------------------------
extern "C" void kernel_launch(void* const* d_in, const int* in_sizes, int n_in,
                              void* d_out, int out_size, void* d_ws, size_t ws_size,
                              hipStream_t stream) {
    const int*   inputs   = (const int*)  d_in[0];
    const int*   masks    = (const int*)  d_in[1];
    const int*   entities = (const int*)  d_in[2];
    const int*   labels   = (const int*)  d_in[3];
    const float* W_word   = (const float*)d_in[4];
    const float* W_ent    = (const float*)d_in[5];
    const float* fc_w_W   = (const float*)d_in[6];
    const float* fc_w_b   = (const float*)d_in[7];
    const float* fc_e_W   = (const float*)d_in[8];
    const float* fc_e_b   = (const float*)d_in[9];
    const float* fc_c_W   = (const float*)d_in[10];
    const float* fc_c_b   = (const float*)d_in[11];
    const float* cls_W    = (const float*)d_in[12];
    const float* cls_b    = (const float*)d_in[13];
    float* out = (float*)d_out;

    // Workspace layout (floats), total ~2.4 MB
    float* segA     = (float*)d_ws;                      // [3*B_, D_]   = 230400
    float* entA     = segA + (size_t)3 * B_ * D_;        // [B_, DE_]    =  25600
    float* features = entA + (size_t)B_ * DE_;           // [B_, CAT_]   = 212992
    float* hidden   = features + (size_t)B_ * CAT_;      // [B_, H_]     = 131072
    float* losses   = hidden + (size_t)B_ * H_;          // [B_]         =    256

    // 1) embedding gather + segment maxes (stacked A for the word GEMM)
    embed_segmax<<<B_, 96, 0, stream>>>(inputs, masks, W_word, segA);

    // 2) entity gather
    ent_gather<<<B_, 128, 0, stream>>>(entities, W_ent, entA);

    // 3) word path GEMM: [768,300] x [300,256] -> scatter to features cols [0,768)
    gemm_wmma_gelu<0><<<dim3(HW_ / 32, (3 * B_) / 16), 32, 0, stream>>>(
        segA, D_, fc_w_W, HW_, fc_w_b, features, D_);

    // 4) entity path GEMM: [256,100] x [100,64] -> features cols [768,832)
    gemm_wmma_gelu<1><<<dim3(HE_ / 32, B_ / 16), 32, 0, stream>>>(
        entA, DE_, fc_e_W, HE_, fc_e_b, features, DE_);

    // 5) combine GEMM: [256,832] x [832,512] -> hidden
    gemm_wmma_gelu<2><<<dim3(H_ / 32, B_ / 16), 32, 0, stream>>>(
        features, CAT_, fc_c_W, H_, fc_c_b, hidden, CAT_);

    // 6) classifier + per-sample loss
    cls_loss<<<B_, 64, 0, stream>>>(hidden, cls_W, cls_b, labels, out, losses);

    // 7) mean loss -> out[512]
    loss_mean<<<1, B_, 0, stream>>>(losses, out);
}